// MLoRAFCN_33423435497722
// MI455X (gfx1250) — compile-verified
//
#include <hip/hip_runtime.h>

typedef __attribute__((ext_vector_type(2))) float v2f;
typedef __attribute__((ext_vector_type(8))) float v8f;

#define B_ROWS 32768
#define D_DIM  1024
#define U_DIM  1024
#define R_DIM  4
#define H_DIM  16
#define NDOM   10   // active domains 1..10

// workspace layout (float element offsets)
#define WS_DOM   0                         // B*4
#define WS_BEFF  (B_ROWS * 4)              // 4*U
#define WS_DBIAS (WS_BEFF + 4 * U_DIM)     // U
#define WS_M4    (WS_DBIAS + U_DIM)        // 16
#define WS_VL    (WS_M4 + 16)              // 16
#define WS_WMIX  (WS_VL + 16)              // 16
#define WS_AALL  (WS_WMIX + 16)            // D*40

// ---------------------------------------------------------------------------
// K1: tiny prep — softmax mixture, Beff, dbias, M4 = WQ WK^T/sqrt(H), VL = WV·linear,
//     packed Aall[d][n*4+r]
// ---------------------------------------------------------------------------
__global__ __launch_bounds__(256) void prep_kernel(
    const int* __restrict__ dind, const float* __restrict__ a_kernel,
    const float* __restrict__ b_kernel, const float* __restrict__ domain_bias,
    const float* __restrict__ WQ, const float* __restrict__ WK,
    const float* __restrict__ WV, const float* __restrict__ linear,
    const float* __restrict__ factor, float* __restrict__ ws)
{
    __shared__ float wsm[NDOM];
    const int tid = threadIdx.x;

    if (tid == 0) {
        // softmax over all ND=11 logits, keep cols 1..10
        float f[11], mx = -1e30f;
        for (int i = 0; i < 11; i++) { f[i] = factor[i]; mx = fmaxf(mx, f[i]); }
        float s = 0.f;
        for (int i = 0; i < 11; i++) { f[i] = __expf(f[i] - mx); s += f[i]; }
        const float inv = 1.f / s;
        for (int n = 0; n < NDOM; n++) { wsm[n] = f[1 + n] * inv; ws[WS_WMIX + n] = wsm[n]; }
        // M4[i][j] = (1/sqrt(H)) * sum_h WQ[i,h]*WK[j,h];  VL[i][j] = sum_h WV[i,h]*linear[h,j]
        for (int i = 0; i < 4; i++)
            for (int j = 0; j < 4; j++) {
                float aqk = 0.f, avl = 0.f;
                for (int h = 0; h < H_DIM; h++) {
                    aqk += WQ[i * H_DIM + h] * WK[j * H_DIM + h];
                    avl += WV[i * H_DIM + h] * linear[h * R_DIM + j];
                }
                ws[WS_M4 + i * 4 + j] = aqk * 0.25f;   // 1/sqrt(16)
                ws[WS_VL + i * 4 + j] = avl;
            }
    }
    // dbias = domain_bias[domain_indicator[0,0]]
    const int idx = dind[0];
    for (int u = tid; u < U_DIM; u += 256)
        ws[WS_DBIAS + u] = domain_bias[(size_t)idx * U_DIM + u];
    // Aall[d][n*4+r] = a_kernel[1+n][d][r]
    for (int e = tid; e < D_DIM * 40; e += 256) {
        const int d = e / 40, j = e % 40, n = j >> 2, r = j & 3;
        ws[WS_AALL + e] = a_kernel[(size_t)(1 + n) * D_DIM * R_DIM + d * R_DIM + r];
    }
    __syncthreads();
    // Beff[r][u] = sum_n w[n]*b_kernel[1+n][r][u]
    for (int e = tid; e < 4 * U_DIM; e += 256) {
        const int r = e / U_DIM, u = e % U_DIM;
        float acc = 0.f;
        for (int n = 0; n < NDOM; n++)
            acc += wsm[n] * b_kernel[(size_t)(1 + n) * R_DIM * U_DIM + r * U_DIM + u];
        ws[WS_BEFF + e] = acc;
    }
}

// ---------------------------------------------------------------------------
// K2: dom[B,4] — dA = x@Aall (32 rows/WG, LDS-chunked), then per-row attention
//     collapsed to 4x4 forms: S = dA M4 dA^T; dom = sum_m (sum_n A[n,m]) (dA VL)[m]
// ---------------------------------------------------------------------------
__global__ __launch_bounds__(256) void dom_kernel(const float* __restrict__ x,
                                                  float* __restrict__ ws)
{
    __shared__ float xs[32 * 128];     // 16 KB
    __shared__ float as[128 * 40];     // 20 KB
    __shared__ float dAsh[32 * 40];    //  5 KB
    const int tid = threadIdx.x;
    const int rowblk = blockIdx.x * 32;
    const float* __restrict__ aall = ws + WS_AALL;

    const int row = tid >> 3;   // 0..31
    const int jg  = tid & 7;    // 0..7 -> columns jg*5 .. jg*5+4
    float part[5] = {0.f, 0.f, 0.f, 0.f, 0.f};

    for (int k0 = 0; k0 < D_DIM; k0 += 128) {
        __syncthreads();
        for (int e = tid * 4; e < 32 * 128; e += 1024) {
            const int r = e >> 7, c = e & 127;
            *(float4*)&xs[e] = *(const float4*)&x[(size_t)(rowblk + r) * D_DIM + k0 + c];
        }
        for (int e = tid * 4; e < 128 * 40; e += 1024) {
            *(float4*)&as[e] = *(const float4*)&aall[(size_t)k0 * 40 + e];
        }
        __syncthreads();
        for (int d = 0; d < 128; d++) {
            const float xv = xs[row * 128 + d];
            const float* ap = &as[d * 40 + jg * 5];
            part[0] += xv * ap[0]; part[1] += xv * ap[1]; part[2] += xv * ap[2];
            part[3] += xv * ap[3]; part[4] += xv * ap[4];
        }
    }
    for (int q = 0; q < 5; q++) dAsh[row * 40 + jg * 5 + q] = part[q];
    __syncthreads();

    if (tid < 32) {
        float dA[NDOM][4];
        for (int n = 0; n < NDOM; n++)
            for (int r = 0; r < 4; r++) dA[n][r] = dAsh[tid * 40 + n * 4 + r];
        float M4[16], VL[16];
        for (int i = 0; i < 16; i++) { M4[i] = ws[WS_M4 + i]; VL[i] = ws[WS_VL + i]; }
        // t = dA @ M4
        float t[NDOM][4];
        for (int n = 0; n < NDOM; n++)
            for (int r = 0; r < 4; r++) {
                float a = 0.f;
                for (int rr = 0; rr < 4; rr++) a += dA[n][rr] * M4[rr * 4 + r];
                t[n][r] = a;
            }
        // per-row softmax of S[n][m] = t[n]·dA[m]; accumulate column sums c[m]
        float c[NDOM];
        for (int m = 0; m < NDOM; m++) c[m] = 0.f;
        for (int n = 0; n < NDOM; n++) {
            float srow[NDOM], mx = -1e30f;
            for (int m = 0; m < NDOM; m++) {
                float s = 0.f;
                for (int r = 0; r < 4; r++) s += t[n][r] * dA[m][r];
                srow[m] = s; mx = fmaxf(mx, s);
            }
            float ssum = 0.f;
            for (int m = 0; m < NDOM; m++) { srow[m] = __expf(srow[m] - mx); ssum += srow[m]; }
            const float inv = 1.f / ssum;
            for (int m = 0; m < NDOM; m++) c[m] += srow[m] * inv;
        }
        // dom = sum_m c[m] * (dA[m] @ VL)
        float d0 = 0.f, d1 = 0.f, d2 = 0.f, d3 = 0.f;
        for (int m = 0; m < NDOM; m++) {
            float u0 = 0.f, u1 = 0.f, u2 = 0.f, u3 = 0.f;
            for (int rr = 0; rr < 4; rr++) {
                const float a = dA[m][rr];
                u0 += a * VL[rr * 4 + 0]; u1 += a * VL[rr * 4 + 1];
                u2 += a * VL[rr * 4 + 2]; u3 += a * VL[rr * 4 + 3];
            }
            d0 += c[m] * u0; d1 += c[m] * u1; d2 += c[m] * u2; d3 += c[m] * u3;
        }
        const int grow = rowblk + tid;
        ws[WS_DOM + grow * 4 + 0] = d0;
        ws[WS_DOM + grow * 4 + 1] = d1;
        ws[WS_DOM + grow * 4 + 2] = d2;
        ws[WS_DOM + grow * 4 + 3] = d3;
    }
}

// ---------------------------------------------------------------------------
// K3: out = relu(x@W + bias + dom@Beff + dbias) with v_wmma_f32_16x16x4_f32
//     WG = 256 threads (8 wave32, 4(M)x2(N)), tile 128x256, K-chunks of 32.
//     W tile staged TRANSPOSED in LDS ([N][KC+2] pad) so every B operand is a
//     contiguous 8B pair -> single ds_load_b64, no VGPR shuffle movs, and the
//     stride-34 layout spreads the 16 active lanes over 16 distinct banks.
// ---------------------------------------------------------------------------
#define M_BLK 128
#define N_BLK 256
#define KC    32
#define LDW   (KC + 2)   // padded transposed-W row stride (dwords)

__global__ __launch_bounds__(256) void gemm_kernel(
    const float* __restrict__ x, const float* __restrict__ W,
    const float* __restrict__ bias, const float* __restrict__ ws,
    float* __restrict__ out)
{
    __shared__ float xsh[M_BLK * KC];     // 16 KB
    __shared__ float wshT[N_BLK * LDW];   // 34 KB, transposed W tile

    const int tid  = threadIdx.x;
    const int mb   = blockIdx.y * M_BLK;
    const int nb   = blockIdx.x * N_BLK;
    const int wave = tid >> 5;          // wave32
    const int lane = tid & 31;
    const int wm   = wave >> 1;         // 0..3 -> rows wm*32
    const int wn   = wave & 1;          // 0..1 -> cols wn*128
    const int lh   = lane >> 4;         // half-wave: K pair select / M+8 select
    const int lc   = lane & 15;         // A row / B,C column within tile
    const int koff = lh * 2;            // lanes 0-15: K=0,1 ; lanes 16-31: K=2,3

    v8f acc[2][8] = {};

    for (int k0 = 0; k0 < D_DIM; k0 += KC) {
        __syncthreads();
        // stage x tile [M_BLK][KC]
        for (int e = tid * 4; e < M_BLK * KC; e += 1024) {
            const int r = e >> 5, c = e & (KC - 1);
            *(float4*)&xsh[e] = *(const float4*)&x[(size_t)(mb + r) * D_DIM + k0 + c];
        }
        // stage W tile transposed: wshT[n][k] = W[k0+k][nb+n]
        for (int e = tid * 4; e < KC * N_BLK; e += 1024) {
            const int r = e >> 8, c = e & (N_BLK - 1);
            const float4 wv = *(const float4*)&W[(size_t)(k0 + r) * U_DIM + nb + c];
            wshT[(c + 0) * LDW + r] = wv.x;
            wshT[(c + 1) * LDW + r] = wv.y;
            wshT[(c + 2) * LDW + r] = wv.z;
            wshT[(c + 3) * LDW + r] = wv.w;
        }
        // prefetch next W chunk into L2 while we compute
        if (k0 + KC < D_DIM) {
            __builtin_prefetch(&W[(size_t)(k0 + KC + (tid >> 3)) * U_DIM + nb + (tid & 7) * 32], 0, 0);
        }
        __syncthreads();

        for (int kk = 0; kk < KC; kk += 4) {
            // A operands (16x4 f32): lane holds row (lc), K-pair per half-wave
            const v2f a0 = *(const v2f*)&xsh[(wm * 32 + lc) * KC + kk + koff];
            const v2f a1 = *(const v2f*)&xsh[(wm * 32 + 16 + lc) * KC + kk + koff];
#pragma unroll
            for (int t = 0; t < 8; t++) {
                const int col = wn * 128 + t * 16 + lc;
                // B operand (4x16 f32): contiguous K-pair in transposed LDS tile
                const v2f bv = *(const v2f*)&wshT[col * LDW + kk + koff];
                acc[0][t] = __builtin_amdgcn_wmma_f32_16x16x4_f32(
                    false, a0, false, bv, (short)0, acc[0][t], false, false);
                acc[1][t] = __builtin_amdgcn_wmma_f32_16x16x4_f32(
                    false, a1, false, bv, (short)0, acc[1][t], false, false);
            }
        }
    }

    // epilogue: + bias + dom@Beff + dbias, relu
    const float* __restrict__ dom   = ws + WS_DOM;
    const float* __restrict__ Beff  = ws + WS_BEFF;
    const float* __restrict__ dbias = ws + WS_DBIAS;

    float4 dmv[2][8];
#pragma unroll
    for (int mt = 0; mt < 2; mt++)
#pragma unroll
        for (int i = 0; i < 8; i++) {
            const int m = mb + wm * 32 + mt * 16 + i + 8 * lh;
            dmv[mt][i] = *(const float4*)&dom[(size_t)m * 4];
        }

#pragma unroll
    for (int t = 0; t < 8; t++) {
        const int n = nb + wn * 128 + t * 16 + lc;
        const float b0 = Beff[n], b1 = Beff[U_DIM + n];
        const float b2 = Beff[2 * U_DIM + n], b3 = Beff[3 * U_DIM + n];
        const float bb = bias[n] + dbias[n];
#pragma unroll
        for (int mt = 0; mt < 2; mt++)
#pragma unroll
            for (int i = 0; i < 8; i++) {
                const int m = mb + wm * 32 + mt * 16 + i + 8 * lh;
                const float4 dm = dmv[mt][i];
                float v = acc[mt][t][i] + bb
                        + dm.x * b0 + dm.y * b1 + dm.z * b2 + dm.w * b3;
                out[(size_t)m * U_DIM + n] = fmaxf(v, 0.f);
            }
    }
}

// ---------------------------------------------------------------------------
extern "C" void kernel_launch(void* const* d_in, const int* in_sizes, int n_in,
                              void* d_out, int out_size, void* d_ws, size_t ws_size,
                              hipStream_t stream) {
    const float* x           = (const float*)d_in[0];
    const int*   dind        = (const int*)d_in[1];
    const float* W           = (const float*)d_in[2];
    const float* bias        = (const float*)d_in[3];
    const float* a_kernel    = (const float*)d_in[4];
    const float* b_kernel    = (const float*)d_in[5];
    const float* domain_bias = (const float*)d_in[6];
    const float* WQ          = (const float*)d_in[7];
    const float* WK          = (const float*)d_in[8];
    const float* WV          = (const float*)d_in[9];
    const float* linear      = (const float*)d_in[10];
    const float* factor      = (const float*)d_in[11];
    float* ws  = (float*)d_ws;
    float* out = (float*)d_out;

    hipLaunchKernelGGL(prep_kernel, dim3(1), dim3(256), 0, stream,
                       dind, a_kernel, b_kernel, domain_bias, WQ, WK, WV, linear, factor, ws);
    hipLaunchKernelGGL(dom_kernel, dim3(B_ROWS / 32), dim3(256), 0, stream, x, ws);
    hipLaunchKernelGGL(gemm_kernel, dim3(U_DIM / N_BLK, B_ROWS / M_BLK), dim3(256), 0, stream,
                       x, W, bias, ws, out);
}